// A2C_6571299963318
// MI455X (gfx1250) — compile-verified
//
#include <hip/hip_runtime.h>
#include <math.h>

// ---------------------------------------------------------------------------
// Types for WMMA f32 16x16x4:  A is 16x4 (2 f32/lane), B is 4x16 (2 f32/lane),
// C/D is 16x16 f32 (8 f32/lane).
// ---------------------------------------------------------------------------
typedef __attribute__((ext_vector_type(2))) float v2f;
typedef __attribute__((ext_vector_type(8))) float v8f;
typedef __attribute__((ext_vector_type(4))) int   v4i;

#define N_CH 128   // in_channels C
#define H_CH 32    // hidden width

#if __has_builtin(__builtin_amdgcn_global_load_async_to_lds_b128) && \
    __has_builtin(__builtin_amdgcn_s_wait_asynccnt)
#define USE_ASYNC_LDS 1
typedef __attribute__((address_space(1))) v4i* gv4i_ptr;   // global (AS1)
typedef __attribute__((address_space(3))) v4i* lv4i_ptr;   // LDS    (AS3)
#else
#define USE_ASYNC_LDS 0
#endif

__device__ __forceinline__ float softplus_stable(float x) {
    // jax.nn.softplus: max(x,0) + log1p(exp(-|x|))
    return fmaxf(x, 0.0f) + log1pf(expf(-fabsf(x)));
}

// ---------------------------------------------------------------------------
// Degree: deg[dst] += 1 per edge  (self-loop +1 folded into rsqrt pass)
// ---------------------------------------------------------------------------
__global__ void degree_kernel(const int* __restrict__ dst, float* __restrict__ deg, int E) {
    int e = blockIdx.x * blockDim.x + threadIdx.x;
    if (e < E) atomicAdd(&deg[dst[e]], 1.0f);
}

__global__ void dinv_kernel(float* __restrict__ deg, int n) {
    int i = blockIdx.x * blockDim.x + threadIdx.x;
    if (i < n) deg[i] = rsqrtf(deg[i] + 1.0f);  // +1 for self-loop; always > 0
}

// ---------------------------------------------------------------------------
// GEMM D[M,Nc] = A[M,K] @ B[K,Nc] (+bias, optional ReLU) via V_WMMA_F32_16X16X4_F32.
//
// - Whole B matrix staged into LDS per block (max 64 KB << 320 KB WGP LDS),
//   via gfx1250 async global->LDS (ASYNCcnt path).
// - One wave computes NTILE adjacent 16x16 output tiles along N, so each
//   A-fragment load feeds NTILE WMMAs.
// - M is a multiple of 16 (100000 = 6250*16). tilesN must be divisible by NTILE.
//
// f32 A-matrix 16x4 layout (ISA 7.12.2): lanes 0-15 row M=lane, v0=K0,v1=K1;
// lanes 16-31 same rows, v0=K2,v1=K3. B mirrors with N=lane&15.
// C/D: vgpr r holds row (r + 8*(lane>>4)), col (lane&15).
// ---------------------------------------------------------------------------
template <int RELU, int NTILE>
__global__ void gemm_wmma_f32(const float* __restrict__ A, const float* __restrict__ B,
                              const float* __restrict__ bias, float* __restrict__ D,
                              int M, int K, int Nc) {
    extern __shared__ float Blds[];  // K * Nc floats

    // ---- cooperative staging of B into LDS (all threads, before any return) ----
    {
        const int nvec = (K * Nc) >> 2;  // float4 count
#if USE_ASYNC_LDS
        for (int v = threadIdx.x; v < nvec; v += blockDim.x) {
            __builtin_amdgcn_global_load_async_to_lds_b128(
                (gv4i_ptr)(const_cast<float*>(B) + 4 * v),
                (lv4i_ptr)(Blds + 4 * v),
                /*offset=*/0, /*cpol=*/0);
        }
        __builtin_amdgcn_s_wait_asynccnt(0);
#else
        for (int v = threadIdx.x; v < nvec; v += blockDim.x)
            ((float4*)Blds)[v] = ((const float4*)B)[v];
#endif
        __syncthreads();
    }

    const int wave   = (blockIdx.x * blockDim.x + threadIdx.x) >> 5;
    const int lane   = threadIdx.x & 31;
    const int tilesM = M >> 4;
    const int wavesN = (Nc >> 4) / NTILE;
    const int tm  = wave / wavesN;
    const int twn = wave - tm * wavesN;
    if (tm >= tilesM) return;               // wave-uniform: EXEC all-ones at WMMA

    const int half = lane >> 4;             // 0: K pair {0,1}; 1: K pair {2,3}
    const int idx  = lane & 15;
    const int arow = tm * 16 + idx;         // A row handled by this lane

    int bcol[NTILE];
#pragma unroll
    for (int j = 0; j < NTILE; ++j) bcol[j] = (twn * NTILE + j) * 16 + idx;

    const float* __restrict__ Ar = A + (size_t)arow * K;

    v8f acc[NTILE];
#pragma unroll
    for (int j = 0; j < NTILE; ++j) acc[j] = (v8f){};

    for (int k0 = 0; k0 < K; k0 += 4) {
        const int ka = k0 + 2 * half;
        v2f a; a.x = Ar[ka];
               a.y = Ar[ka + 1];
#pragma unroll
        for (int j = 0; j < NTILE; ++j) {
            v2f b; b.x = Blds[ka * Nc + bcol[j]];
                   b.y = Blds[(ka + 1) * Nc + bcol[j]];
            acc[j] = __builtin_amdgcn_wmma_f32_16x16x4_f32(
                /*neg_a=*/false, a, /*neg_b=*/false, b,
                /*c_mod=*/(short)0, acc[j], /*reuse_a=*/false, /*reuse_b=*/false);
        }
    }

#pragma unroll
    for (int j = 0; j < NTILE; ++j) {
        const float bv = bias ? bias[bcol[j]] : 0.0f;
#pragma unroll
        for (int r = 0; r < 8; ++r) {
            const int orow = tm * 16 + r + 8 * half;
            float v = acc[j][r] + bv;
            if (RELU) v = fmaxf(v, 0.0f);
            D[(size_t)orow * Nc + bcol[j]] = v;
        }
    }
}

// ---------------------------------------------------------------------------
// Edge scatter: agg[dst] += h[src] * dinv[src]*dinv[dst].
// One thread per (edge, 4-channel group): global_load_b128 + 4 f32 atomics.
// h and agg are L2-resident (51.2 MB each vs 192 MB L2) on MI455X.
// ---------------------------------------------------------------------------
__global__ void scatter_edges(const int* __restrict__ src, const int* __restrict__ dst,
                              const float* __restrict__ h, const float* __restrict__ dinv,
                              float* __restrict__ agg, long long total /* E*32 */) {
    long long t = (long long)blockIdx.x * blockDim.x + threadIdx.x;
    if (t >= total) return;
    const int e = (int)(t >> 5);
    const int g = (int)(t & 31);            // 4-channel group
    const int s = src[e];
    const int d = dst[e];
    const float w = dinv[s] * dinv[d];
    const float4 h4 = ((const float4*)h)[(size_t)s * (N_CH / 4) + g];
    float* base = agg + (size_t)d * N_CH + g * 4;
    atomicAdd(base + 0, h4.x * w);
    atomicAdd(base + 1, h4.y * w);
    atomicAdd(base + 2, h4.z * w);
    atomicAdd(base + 3, h4.w * w);
}

// ---------------------------------------------------------------------------
// Fused: self-loop + conv bias + ReLU + residual, in place into agg (float4).
//   agg = relu(agg + h*dinv^2 + b) + x
// ---------------------------------------------------------------------------
__global__ void postconv_kernel(float* __restrict__ agg, const float* __restrict__ h,
                                const float* __restrict__ dinv, const float* __restrict__ bias,
                                const float* __restrict__ x, long long total /* n*32 */) {
    long long t = (long long)blockIdx.x * blockDim.x + threadIdx.x;
    if (t >= total) return;
    const int i = (int)(t >> 5);
    const int g = (int)(t & 31);
    const float di = dinv[i];
    const float w  = di * di;
    float4       a4 = ((float4*)agg)[t];
    const float4 h4 = ((const float4*)h)[t];
    const float4 x4 = ((const float4*)x)[t];
    const float4 b4 = ((const float4*)bias)[g];
    a4.x = fmaxf(a4.x + h4.x * w + b4.x, 0.0f) + x4.x;
    a4.y = fmaxf(a4.y + h4.y * w + b4.y, 0.0f) + x4.y;
    a4.z = fmaxf(a4.z + h4.z * w + b4.z, 0.0f) + x4.z;
    a4.w = fmaxf(a4.w + h4.w * w + b4.w, 0.0f) + x4.w;
    ((float4*)agg)[t] = a4;
}

// ---------------------------------------------------------------------------
// Actor head: [N,32] @ [32,4] + b, softplus -> concentration [N], taylor [N,3]
// ---------------------------------------------------------------------------
__global__ void actor_head(const float* __restrict__ a2, const float* __restrict__ W3,
                           const float* __restrict__ b3, float* __restrict__ conc,
                           float* __restrict__ taylor, int n) {
    int i = blockIdx.x * blockDim.x + threadIdx.x;
    if (i >= n) return;
    const float* ar = a2 + (size_t)i * H_CH;
    float o[4];
#pragma unroll
    for (int j = 0; j < 4; ++j) {
        float s = b3[j];
        for (int k = 0; k < H_CH; ++k) s += ar[k] * W3[k * 4 + j];
        o[j] = s;
    }
    conc[i]           = softplus_stable(o[0]) + 1e-20f;
    taylor[i * 3 + 0] = softplus_stable(o[1]) + 1e-20f;
    taylor[i * 3 + 1] = softplus_stable(o[2]) + 1e-20f;
    taylor[i * 3 + 2] = softplus_stable(o[3]) + 1e-20f;
}

// ---------------------------------------------------------------------------
// Critic readout: csum[c] = sum_i act[i,c]   (per-block partials, 128 atomics/block)
// ---------------------------------------------------------------------------
__global__ void colsum_kernel(const float* __restrict__ act, float* __restrict__ csum, int n) {
    const int c = threadIdx.x;  // blockDim.x == 128
    float s = 0.0f;
    for (int i = blockIdx.x; i < n; i += gridDim.x)
        s += act[(size_t)i * N_CH + c];
    atomicAdd(&csum[c], s);
}

// ---------------------------------------------------------------------------
// Critic MLP on the single readout vector: 128 -> 32 -> 32 -> 1
// ---------------------------------------------------------------------------
__global__ void critic_mlp(const float* __restrict__ csum,
                           const float* __restrict__ W1, const float* __restrict__ b1,
                           const float* __restrict__ W2, const float* __restrict__ b2,
                           const float* __restrict__ W3, const float* __restrict__ b3,
                           float* __restrict__ value) {
    __shared__ float v[N_CH];
    __shared__ float h1[H_CH];
    __shared__ float h2[H_CH];
    const int t = threadIdx.x;  // blockDim.x == 128
    v[t] = csum[t];
    __syncthreads();
    if (t < H_CH) {
        float s = b1[t];
        for (int k = 0; k < N_CH; ++k) s += v[k] * W1[k * H_CH + t];
        h1[t] = fmaxf(s, 0.0f);
    }
    __syncthreads();
    if (t < H_CH) {
        float s = b2[t];
        for (int k = 0; k < H_CH; ++k) s += h1[k] * W2[k * H_CH + t];
        h2[t] = fmaxf(s, 0.0f);
    }
    __syncthreads();
    if (t == 0) {
        float s = b3[0];
        for (int k = 0; k < H_CH; ++k) s += h2[k] * W3[k];
        value[0] = s;
    }
}

// ---------------------------------------------------------------------------
// Host orchestration
// ---------------------------------------------------------------------------
static inline int cdiv(long long a, long long b) { return (int)((a + b - 1) / b); }

static void run_gcn_branch(const float* x, const int* src, const int* dst,
                           const float* convW, const float* convB, const float* dinv,
                           float* hbuf, float* agg, int n, int E, hipStream_t stream) {
    // h = x @ W   (WMMA f32 GEMM, 16x64 strip per wave, B staged in LDS)
    {
        long long waves = (long long)(n >> 4) * ((N_CH >> 4) / 4);
        size_t shmem = (size_t)N_CH * N_CH * sizeof(float);   // 64 KB
        gemm_wmma_f32<0, 4><<<cdiv(waves * 32, 256), 256, shmem, stream>>>(
            x, convW, nullptr, hbuf, n, N_CH, N_CH);
    }
    (void)hipMemsetAsync(agg, 0, (size_t)n * N_CH * sizeof(float), stream);
    {
        long long total = (long long)E * (N_CH / 4);
        scatter_edges<<<cdiv(total, 256), 256, 0, stream>>>(src, dst, hbuf, dinv, agg, total);
    }
    {
        long long total = (long long)n * (N_CH / 4);
        postconv_kernel<<<cdiv(total, 256), 256, 0, stream>>>(agg, hbuf, dinv, convB, x, total);
    }
}

extern "C" void kernel_launch(void* const* d_in, const int* in_sizes, int n_in,
                              void* d_out, int out_size, void* d_ws, size_t ws_size,
                              hipStream_t stream) {
    (void)n_in; (void)out_size; (void)ws_size;

    const float* x    = (const float*)d_in[0];
    const int*   eidx = (const int*)d_in[1];
    const float* aW   = (const float*)d_in[2];
    const float* ab   = (const float*)d_in[3];
    const float* aW1  = (const float*)d_in[4];
    const float* ab1  = (const float*)d_in[5];
    const float* aW2  = (const float*)d_in[6];
    const float* ab2  = (const float*)d_in[7];
    const float* aW3  = (const float*)d_in[8];
    const float* ab3  = (const float*)d_in[9];
    const float* cW   = (const float*)d_in[10];
    const float* cb   = (const float*)d_in[11];
    const float* cW1  = (const float*)d_in[12];
    const float* cb1  = (const float*)d_in[13];
    const float* cW2  = (const float*)d_in[14];
    const float* cb2  = (const float*)d_in[15];
    const float* cW3  = (const float*)d_in[16];
    const float* cb3  = (const float*)d_in[17];

    const int n = in_sizes[0] / N_CH;       // 100000
    const int E = in_sizes[1] / 2;          // 1600000
    const int* src = eidx;
    const int* dst = eidx + E;

    // Workspace (floats): hbuf[n*128] | agg[n*128] | m1[n*32] | m2[n*32] | dinv[n] | csum[128]
    float* hbuf = (float*)d_ws;
    float* agg  = hbuf + (size_t)n * N_CH;
    float* m1   = agg  + (size_t)n * N_CH;
    float* m2   = m1   + (size_t)n * H_CH;
    float* dinv = m2   + (size_t)n * H_CH;
    float* csum = dinv + n;

    float* out_conc   = (float*)d_out;            // [n]
    float* out_taylor = out_conc + n;             // [n,3]
    float* out_value  = out_conc + (size_t)4 * n; // [1]

    // ---- degree / normalization ----
    (void)hipMemsetAsync(dinv, 0, (size_t)n * sizeof(float), stream);
    degree_kernel<<<cdiv(E, 256), 256, 0, stream>>>(dst, dinv, E);
    dinv_kernel<<<cdiv(n, 256), 256, 0, stream>>>(dinv, n);

    // ---- actor branch ----
    run_gcn_branch(x, src, dst, aW, ab, dinv, hbuf, agg, n, E, stream);
    {   // l1: [n,128] @ [128,32] + b, relu   (16x32 strip per wave)
        long long waves = (long long)(n >> 4) * ((H_CH >> 4) / 2);
        gemm_wmma_f32<1, 2><<<cdiv(waves * 32, 256), 256,
                              (size_t)N_CH * H_CH * sizeof(float), stream>>>(
            agg, aW1, ab1, m1, n, N_CH, H_CH);
        // l2: [n,32] @ [32,32] + b, relu
        gemm_wmma_f32<1, 2><<<cdiv(waves * 32, 256), 256,
                              (size_t)H_CH * H_CH * sizeof(float), stream>>>(
            m1, aW2, ab2, m2, n, H_CH, H_CH);
    }
    actor_head<<<cdiv(n, 256), 256, 0, stream>>>(m2, aW3, ab3, out_conc, out_taylor, n);

    // ---- critic branch ----
    run_gcn_branch(x, src, dst, cW, cb, dinv, hbuf, agg, n, E, stream);
    (void)hipMemsetAsync(csum, 0, N_CH * sizeof(float), stream);
    colsum_kernel<<<512, N_CH, 0, stream>>>(agg, csum, n);
    critic_mlp<<<1, N_CH, 0, stream>>>(csum, cW1, cb1, cW2, cb2, cW3, cb3, out_value);
}